// TransR_37967510897466
// MI455X (gfx1250) — compile-verified
//
#include <hip/hip_runtime.h>
#include <hip/hip_bf16.h>

// TransR-style edge scoring:
//   d[e] = || P[row] - P[col] + g[type] ||_1   with  P = H W^T   (bias cancels)
//
// Phase 1: P = H(100000x128) * W^T(128x128) via V_WMMA_F32_16X16X4_F32 (full f32).
//   - H tile staged to LDS with GLOBAL_LOAD_ASYNC_TO_LDS_B128 (ASYNCcnt path)
//   - W staged as K-pair-interleaved LDS so one ds_load_b64 = one B fragment
// Phase 2: per-edge L1 gather-reduce; P (51.2MB) is L2-resident on MI455X (192MB L2).

typedef __attribute__((ext_vector_type(2))) float v2f;
typedef __attribute__((ext_vector_type(8))) float v8f;

#define DIM 128
// W pair layout: row kp holds (W[n][2kp], W[n][2kp+1]) interleaved over n:
// 256 payload dwords padded to 288 (288 % 64 == 32 -> the two lane-halves of a
// B-fragment ds_load_b64 hit disjoint 32-bank halves).
#define LDP 288
// H tile row stride: 132 % 64 == 4 -> A-fragment ds_load_b64 banks are
// {4i,4i+1} for lanes 0-15 and {4i+2,4i+3} for lanes 16-31: conflict-free.
#define LDH 132

// ---------------------------------------------------------------------------
// Kernel 1: P = H * W^T
// Block = 256 threads = 8 waves; each block computes 128 rows of P.
// ---------------------------------------------------------------------------
__global__ __launch_bounds__(256)
void transr_gemm_kernel(const float* __restrict__ H,
                        const float* __restrict__ W,
                        float* __restrict__ P,
                        int n_nodes) {
    __shared__ float WtP[(DIM / 2) * LDP];   // 72 KB
    __shared__ float Hlds[DIM * LDH];        // 67.6 KB

    const int tid  = threadIdx.x;
    const int row0 = blockIdx.x * 128;

    // ---- Async-stage the 128x128 H tile (memory -> LDS, no VGPR round trip).
    // 4096 16-byte chunks; each thread issues 16 async B128 copies.
    {
        const unsigned hbase = (unsigned)(uintptr_t)Hlds;   // LDS byte offset
        for (int c = tid; c < DIM * (DIM / 4); c += 256) {
            const int r  = c >> 5;            // tile row
            const int kc = (c & 31) << 2;     // col base (4 floats)
            int gr = row0 + r;
            if (gr > n_nodes - 1) gr = n_nodes - 1;   // clamp; stores guarded
            const float* gp = H + (size_t)gr * DIM + kc;
            const unsigned la = hbase + (unsigned)(r * LDH + kc) * 4u;
            asm volatile("global_load_async_to_lds_b128 %0, %1, off"
                         :: "v"(la), "v"(gp) : "memory");
        }
        asm volatile("s_wait_asynccnt 0x0" ::: "memory");
    }

    // ---- Stage W as K-pairs (float4 global reads, 2x ds_store_b64 each).
    {
        const float4* __restrict__ W4 = (const float4*)W;
        for (int c = tid; c < (DIM * DIM) / 4; c += 256) {
            const float4 w = W4[c];
            const int n  = c >> 5;            // W row
            const int k0 = (c & 31) << 2;     // W col base (even)
            const int kp = k0 >> 1;
            const v2f p01 = {w.x, w.y};
            const v2f p23 = {w.z, w.w};
            *(v2f*)&WtP[(kp    ) * LDP + 2 * n] = p01;
            *(v2f*)&WtP[(kp + 1) * LDP + 2 * n] = p23;
        }
    }
    __syncthreads();

    const int wave = tid >> 5;          // 0..7
    const int lane = tid & 31;
    const int half = lane >> 4;         // 0: lanes 0-15, 1: lanes 16-31
    const int l16  = lane & 15;

    const int row_base = row0 + wave * 16;

    // A-fragment source row in LDS (16x4 f32 A: lanes 0-15 K=k0..k0+1,
    // lanes 16-31 K=k0+2..k0+3, M = l16).
    const float* __restrict__ arow = &Hlds[(wave * 16 + l16) * LDH];

    v8f acc[8] = {};   // 8 N-tiles of 16 -> full 128-wide strip

    for (int k0 = 0; k0 < DIM; k0 += 4) {
        const int ka = k0 + half * 2;               // this lane-half's K base

        v2f a = *(const v2f*)(arow + ka);           // one ds_load_b64

        const v2f* __restrict__ wrow = (const v2f*)(WtP + (ka >> 1) * LDP);
#pragma unroll
        for (int j = 0; j < 8; ++j) {
            v2f b = wrow[j * 16 + l16];             // one ds_load_b64 (pairable)

            acc[j] = __builtin_amdgcn_wmma_f32_16x16x4_f32(
                /*neg_a=*/false, a, /*neg_b=*/false, b,
                /*c_mod=*/(short)0, acc[j],
                /*reuse_a=*/false, /*reuse_b=*/false);
        }
    }

    // C/D layout: VGPR v -> M = v (lanes 0-15) or M = 8+v (lanes 16-31), N = l16.
    float* __restrict__ pbase = P + (size_t)(row_base + half * 8) * DIM + l16;

    if (row_base + 16 <= n_nodes) {
        // Full tile: unguarded stores, single address + immediate offsets.
#pragma unroll
        for (int v = 0; v < 8; ++v) {
#pragma unroll
            for (int j = 0; j < 8; ++j) {
                pbase[v * DIM + j * 16] = acc[j][v];
            }
        }
    } else {
        // Partial tile (last block only): guard once per output row.
#pragma unroll
        for (int v = 0; v < 8; ++v) {
            const int m = row_base + half * 8 + v;
            if (m < n_nodes) {
#pragma unroll
                for (int j = 0; j < 8; ++j) {
                    pbase[v * DIM + j * 16] = acc[j][v];
                }
            }
        }
    }
}

// ---------------------------------------------------------------------------
// Kernel 2: per-edge L1 score. One wave32 per edge; one float4 per lane.
// ---------------------------------------------------------------------------
__global__ __launch_bounds__(256)
void transr_edge_kernel(const float* __restrict__ P,
                        const float* __restrict__ G,
                        const int* __restrict__ erow,
                        const int* __restrict__ ecol,
                        const int* __restrict__ etype,
                        float* __restrict__ out,
                        int n_edges) {
    const int wave_id = (int)((blockIdx.x * 256 + threadIdx.x) >> 5);
    const int lane = threadIdx.x & 31;
    if (wave_id >= n_edges) return;

    const int r = erow[wave_id];
    const int c = ecol[wave_id];
    const int t = etype[wave_id];

    const float4* __restrict__ pr = (const float4*)(P + (size_t)r * DIM);
    const float4* __restrict__ pc = (const float4*)(P + (size_t)c * DIM);
    const float4* __restrict__ pg = (const float4*)(G + (size_t)t * DIM);

    const float4 a = pr[lane];
    const float4 b = pc[lane];
    const float4 g = pg[lane];

    float s = fabsf(a.x - b.x + g.x)
            + fabsf(a.y - b.y + g.y)
            + fabsf(a.z - b.z + g.z)
            + fabsf(a.w - b.w + g.w);

    // wave32 reduction
#pragma unroll
    for (int off = 16; off > 0; off >>= 1) {
        s += __shfl_xor(s, off, 32);
    }

    if (lane == 0) out[wave_id] = s;
}

// ---------------------------------------------------------------------------
// Launch
// Inputs (setup_inputs order): h[100000*128] f32, g[500*128] f32,
// edge_idx[2*E] int32, edge_type[E] int32, W[128*128] f32, b[128] f32 (unused:
// bias cancels analytically in s - t).
// ---------------------------------------------------------------------------
extern "C" void kernel_launch(void* const* d_in, const int* in_sizes, int n_in,
                              void* d_out, int out_size, void* d_ws, size_t ws_size,
                              hipStream_t stream) {
    const float* H     = (const float*)d_in[0];
    const float* G     = (const float*)d_in[1];
    const int*   eidx  = (const int*)d_in[2];
    const int*   etype = (const int*)d_in[3];
    const float* W     = (const float*)d_in[4];

    const int n_nodes = in_sizes[0] / DIM;
    const int n_edges = in_sizes[3];

    float* P = (float*)d_ws;   // n_nodes * 128 floats (51.2 MB) of scratch

    const int gemm_blocks = (n_nodes + 127) / 128;
    transr_gemm_kernel<<<gemm_blocks, 256, 0, stream>>>(H, W, P, n_nodes);

    const int edge_blocks = (n_edges + 7) / 8;   // 8 waves per block
    transr_edge_kernel<<<edge_blocks, 256, 0, stream>>>(
        P, G, eidx, eidx + n_edges, etype, (float*)d_out, n_edges);
}